// ExponentialSmoothingNet_31696858644989
// MI455X (gfx1250) — compile-verified
//
#include <hip/hip_runtime.h>

// Exponential smoothing final state: out[b,f] = sum_t w_t * x[b,t,f]
//   w_t = alpha*(1-alpha)^(T-1-t) (t>=1), w_0 = (1-alpha)^(T-1)
// HBM-bound: 256 MB read @ 23.3 TB/s ~ 11 us. Data movement via the Tensor
// Data Mover (tensor_load_to_lds, TENSORcnt, 8KB strided 2-D tiles, double
// buffered => ~16 MB in flight across 2048 waves). Math on the f32 WMMA path;
// LDS consumption through addrspace(3) pointers so it lowers to ds_load.

typedef __attribute__((ext_vector_type(2))) float v2f;
typedef __attribute__((ext_vector_type(8))) float v8f;
typedef __attribute__((ext_vector_type(4))) unsigned int u32x4;
typedef __attribute__((ext_vector_type(8))) int i32x8;
typedef __attribute__((ext_vector_type(4))) int i32x4;
typedef __attribute__((address_space(3))) const float* lds_cfp;

namespace {
constexpr int kB = 128;
constexpr int kT = 2048;
constexpr int kF = 256;
constexpr int kWaves  = 4;             // waves per block (128 threads)
constexpr int kRows   = 128;           // timesteps staged per TDM op (8 KB)
constexpr int kStages = kT / kRows;    // 16
constexpr int kTiles  = kB * (kF / 16);// 2048 (b, f-tile) tiles, 1 wave each
}

#if __has_builtin(__builtin_amdgcn_tensor_load_to_lds)
#define ES_USE_TDM 1
#else
#define ES_USE_TDM 0
#endif

#if ES_USE_TDM
// One TDM op: copy a kRows x 16 f32 tile (row stride kF elements) from global
// to LDS, compacted to 64 B rows. Descriptor per CDNA5 ISA §8.3/8.4.
// 6-arg builtin: (u32x4 g0, i32x8 g1, i32x4 g2, i32x4 g3, i32x8 extra, cpol)
__device__ __forceinline__ void tdm_load_tile(unsigned lds_addr,
                                              const float* gptr) {
  unsigned long long ga = (unsigned long long)(size_t)(const void*)gptr;
  u32x4 g0;
  g0[0] = 1u;                                          // count=1 (valid, user)
  g0[1] = lds_addr;                                    // lds_addr [63:32]
  g0[2] = (unsigned)(ga & 0xFFFFFFFFu);                // global_addr lo
  g0[3] = (unsigned)((ga >> 32) & 0x01FFFFFFu)         // global_addr [56:32]
          | (2u << 30);                                // type=2 ("image")
  i32x8 g1;
  g1[0] = (int)(2u << 16);                 // data_size=2 (4 bytes)
  g1[1] = (int)(16u << 16);                // tensor_dim0 = 16 (lo16 @ [63:48])
  g1[2] = (int)((unsigned)kRows << 16);    // tensor_dim1 = kRows
  g1[3] = (int)(16u << 16);                // tile_dim0 = 16 @ [127:112]
  g1[4] = kRows;                           // tile_dim1 = kRows, tile_dim2 = 0
  g1[5] = kF;                              // tensor_dim0_stride = 256 elems
  g1[6] = 0;                               // stride hi / dim1_stride lo
  g1[7] = 0;                               // dim1_stride hi (unused, 2-D tile)
  i32x4 z4 = {0, 0, 0, 0};                 // groups 2/3 unused (<=2-D)
  i32x8 z8 = {0, 0, 0, 0, 0, 0, 0, 0};
  __builtin_amdgcn_tensor_load_to_lds(g0, g1, z4, z4, z8, 0);
}
#endif

__global__ __launch_bounds__(kWaves * 32, 1)
void es_wmma_tdm_kernel(const float* __restrict__ x,
                        const float* __restrict__ alpha_p,
                        float* __restrict__ out) {
  __shared__ float smem[kWaves][2][kRows * 16];  // 64 KB static LDS

  const int lane = threadIdx.x & 31;
  const int wave = threadIdx.x >> 5;
  const int tile = blockIdx.x * kWaves + wave;   // 0..2047
  const int b    = tile >> 4;
  const int f0   = (tile & 15) << 4;
  const int half = lane >> 4;                    // 0: lanes 0-15, 1: 16-31
  const int fl   = lane & 15;

  const float alpha = alpha_p[0];
  const float q  = 1.0f - alpha;
  const float q2 = q * q;
  const float q3 = q2 * q;
  const float q4 = q2 * q2;
  const float q8 = q4 * q4;

  // Loop-invariant per-lane A-operand selectors (K=0/2 and K=1/3 weights):
  const float c0 = half ? q    : q3;
  const float c1 = half ? 1.0f : q2;

  // Row t of this tile lives at gbase + t*kF (16 floats starting at f0).
  const float* gbase = x + (size_t)b * kT * kF + f0;

  // LDS byte offset of this wave's buffers (low 32 bits of the generic
  // address). Laundered so the compiler (a) treats smem as escaped and
  // (b) can't fold reads of TDM-written memory. Rebuilding an AS(3) pointer
  // from the 32-bit offset keeps the consumption on the ds_load path.
  unsigned lds0 = (unsigned)(size_t)(void*)&smem[wave][0][0];
  asm volatile("" : "+v"(lds0));
  const unsigned lds1 = lds0 + (unsigned)(kRows * 16 * sizeof(float));
  lds_cfp sbuf = (lds_cfp)lds0;

#if ES_USE_TDM
  // Prologue: stages S-1 -> buf0, S-2 -> buf1 (TENSORcnt <= 2 outstanding).
  tdm_load_tile(lds0, gbase + (size_t)(kStages - 1) * kRows * kF);
  tdm_load_tile(lds1, gbase + (size_t)(kStages - 2) * kRows * kF);
#endif

  v8f acc0 = {};
  v8f acc1 = {};
  // u = q^(T-1 - top_of_current_hi_chunk); weight of t is alpha*u*q^(top-t).
  float u = 1.0f;

  for (int s = kStages - 1; s >= 0; --s) {
    const int parity = (kStages - 1 - s) & 1;
    lds_cfp bb = sbuf + parity * (kRows * 16);

#if ES_USE_TDM
    // In-order TDM completion: tensorcnt<=1 means stage s's tile has landed.
    if (s > 0) __builtin_amdgcn_s_wait_tensorcnt(1);
    else       __builtin_amdgcn_s_wait_tensorcnt(0);
    asm volatile("" ::: "memory");
#else
    // Fallback: cooperative direct loads into LDS (b128 per lane).
    {
      float* wb = &smem[wave][parity][0];
      const float* gs = gbase + (size_t)s * kRows * kF;
      for (int i = lane; i < kRows * 4; i += 32) {
        const int row = i >> 2, seg = i & 3;  // 4 lanes per 64 B row
        ((float4*)wb)[i] = ((const float4*)(gs + (size_t)row * kF))[seg];
      }
      asm volatile("s_wait_loadcnt 0" ::: "memory");
    }
#endif

    // Consume kRows high->low, 8 timesteps (2 WMMAs) per iteration.
    const int lbase = half * 32 + fl;   // (half*2)*16 + fl
    for (int rc = kRows - 8; rc >= 0; rc -= 8) {
      lds_cfp phi = bb + (rc + 4) * 16 + lbase;
      lds_cfp plo = bb + rc * 16 + lbase;
      float bh0 = phi[0];
      float bh1 = phi[16];
      float bl0 = plo[0];
      float bl1 = plo[16];

      const float au_hi = alpha * u;   // weight of t_local = rc+7
      const float au_lo = au_hi * q4;  // weight of t_local = rc+3

      v2f A, Bm;
      A[0]  = au_hi * c0;
      A[1]  = au_hi * c1;
      Bm[0] = bh0;
      Bm[1] = bh1;
      acc0 = __builtin_amdgcn_wmma_f32_16x16x4_f32(
          false, A, false, Bm, (short)0, acc0, false, false);

      A[0]  = au_lo * c0;
      A[1]  = au_lo * c1;
      Bm[0] = bl0;
      Bm[1] = bl1;
      acc1 = __builtin_amdgcn_wmma_f32_16x16x4_f32(
          false, A, false, Bm, (short)0, acc1, false, false);

      u *= q8;
    }

#if ES_USE_TDM
    // Refill this buffer with stage s-2 once our DS reads have drained.
    if (s >= 2) {
      asm volatile("s_wait_dscnt 0" ::: "memory");
      tdm_load_tile(parity ? lds1 : lds0,
                    gbase + (size_t)(s - 2) * kRows * kF);
    }
#endif
  }

  // All 16 D rows identical; row M=0 = VGPR 0, lanes 0-15.
  v8f acc = acc0 + acc1;
  // t=0 got weight alpha*q^(T-1) in the loop; want q^(T-1). Correction weight
  // is exactly q^T == u after the final update (also right for alpha=0 / 1).
  if (lane < 16) {
    const float x0 = gbase[lane];  // x[b, 0, f0+lane]
    out[b * kF + f0 + lane] = acc[0] + u * x0;
  }
}

extern "C" void kernel_launch(void* const* d_in, const int* in_sizes, int n_in,
                              void* d_out, int out_size, void* d_ws, size_t ws_size,
                              hipStream_t stream) {
  (void)in_sizes; (void)n_in; (void)out_size; (void)d_ws; (void)ws_size;
  const float* x     = (const float*)d_in[0];
  const float* alpha = (const float*)d_in[1];
  float* out         = (float*)d_out;
  dim3 grid(kTiles / kWaves);   // 512 blocks
  dim3 block(kWaves * 32);      // 128 threads (4 wave32)
  es_wmma_tdm_kernel<<<grid, block, 0, stream>>>(x, alpha, out);
}